// PartitionPtr_70909910057535
// MI455X (gfx1250) — compile-verified
//
#include <hip/hip_runtime.h>
#include <hip/hip_bf16.h>
#include <math.h>

// ---------------- problem constants (match reference) ----------------
constexpr int B   = 32;
constexpr int LD  = 256;
constexpr int LE  = 258;
constexpr int ENC = 512;
constexpr int HID = 512;
constexpr int MLP = 256;
constexpr int NL  = 4;

typedef __attribute__((ext_vector_type(2))) float v2f;
typedef __attribute__((ext_vector_type(8))) float v8f;

// ---------------- fp32 WMMA: D(16x16) = A(16x4) * B(4x16) + C ----------------
// A layout: lane L holds A[L%16][2*(L/16)] , A[L%16][2*(L/16)+1]
// B layout: lane L holds B[2*(L/16)][L%16] , B[2*(L/16)+1][L%16]
// C/D layout: VGPR i: lanes 0-15 -> (M=i, N=lane); lanes 16-31 -> (M=i+8, N=lane-16)
__device__ __forceinline__ v8f wmma_f32(v2f a, v2f b, v8f c) {
    return __builtin_amdgcn_wmma_f32_16x16x4_f32(
        /*neg_a=*/false, a, /*neg_b=*/false, b,
        /*c_mod=*/(short)0, c, /*reuse_a=*/false, /*reuse_b=*/false);
}

__device__ __forceinline__ float sigmoidf_(float x) {
    return 1.0f / (1.0f + __expf(-x));
}

__device__ __forceinline__ v2f ld2(const float* p) {
    v2f r; r.x = p[0]; r.y = p[1]; return r;
}

// ============================================================================
// K1: GI[t][b][0..3H) = d_inputs[b][t][:] @ W_ih^T + b_ih
//     2x2 register-blocked: one wave -> 32x32 output block (4 WMMA / 4 loads).
//     Each 32-row block covers exactly one t and all 32 batch rows.
// ============================================================================
__global__ __launch_bounds__(32) void gi_kernel(
    const float* __restrict__ dx,   // (B, LD, HID)
    const float* __restrict__ Wih,  // (3H, HID) row-major
    const float* __restrict__ bih,  // (3H)
    float* __restrict__ GI)         // (LD, B, 3H)
{
    const int lane = threadIdx.x & 31;
    const int n0 = blockIdx.x * 32;          // output col block (0..3H)
    const int t  = blockIdx.y;               // one t per block (32 rows = batch)
    const int mr = lane & 15;
    const int k0 = (lane >> 4) << 1;

    const float* arow0 = dx + ((size_t)mr * LD + t) * HID;          // b = mr
    const float* arow1 = dx + ((size_t)(16 + mr) * LD + t) * HID;   // b = 16+mr
    const float* brow0 = Wih + (size_t)(n0 + mr) * HID;
    const float* brow1 = Wih + (size_t)(n0 + 16 + mr) * HID;

    v8f c00 = {}, c01 = {}, c10 = {}, c11 = {};
    for (int k = 0; k < HID; k += 4) {
        const v2f a0 = ld2(arow0 + k + k0);
        const v2f a1 = ld2(arow1 + k + k0);
        const v2f b0 = ld2(brow0 + k + k0);
        const v2f b1 = ld2(brow1 + k + k0);
        c00 = wmma_f32(a0, b0, c00);
        c01 = wmma_f32(a0, b1, c01);
        c10 = wmma_f32(a1, b0, c10);
        c11 = wmma_f32(a1, b1, c11);
    }
#pragma unroll
    for (int i = 0; i < 8; ++i) {
        const int row = i + ((lane >> 4) << 3);
        const int col = lane & 15;
        const int n_lo = n0 + col, n_hi = n0 + 16 + col;
        float* g0 = GI + ((size_t)t * B + row) * (3 * HID);
        float* g1 = GI + ((size_t)t * B + 16 + row) * (3 * HID);
        g0[n_lo] = c00[i] + bih[n_lo];
        g0[n_hi] = c01[i] + bih[n_hi];
        g1[n_lo] = c10[i] + bih[n_lo];
        g1[n_hi] = c11[i] + bih[n_hi];
    }
}

// ============================================================================
// K2: one GRU step: gh = h_prev @ W_hh^T (+b_hh), fused gates, writes h_next
//     and d_outputs[:, t, :]. One wave per 16x16 tile of h_new; A fragment
//     shared across the three gate tiles (r/z/n columns j, H+j, 2H+j).
// ============================================================================
__global__ __launch_bounds__(32) void gru_step_kernel(
    const float* __restrict__ hprev,  // (B, HID)
    const float* __restrict__ Whh,    // (3H, HID)
    const float* __restrict__ bhh,    // (3H)
    const float* __restrict__ GI,     // (LD, B, 3H)
    float* __restrict__ hnext,        // (B, HID)
    float* __restrict__ dout,         // (B, LD, HID)
    int t)
{
    const int lane = threadIdx.x & 31;
    const int j0 = blockIdx.x * 16;   // hidden col tile
    const int m0 = blockIdx.y * 16;   // batch row tile
    const int mr = lane & 15;
    const int k0 = (lane >> 4) << 1;

    const float* arow = hprev + (size_t)(m0 + mr) * HID;
    const float* bR = Whh + (size_t)(j0 + mr) * HID;
    const float* bZ = bR + (size_t)HID * HID;
    const float* bN = bZ + (size_t)HID * HID;

    v8f aR = {}, aZ = {}, aN = {};
    for (int k = 0; k < HID; k += 4) {
        const v2f a = ld2(arow + k + k0);
        aR = wmma_f32(a, ld2(bR + k + k0), aR);
        aZ = wmma_f32(a, ld2(bZ + k + k0), aZ);
        aN = wmma_f32(a, ld2(bN + k + k0), aN);
    }

    const float* gi = GI + (size_t)t * B * (3 * HID);
#pragma unroll
    for (int i = 0; i < 8; ++i) {
        const int row = i + ((lane >> 4) << 3);
        const int col = lane & 15;
        const int bidx = m0 + row;
        const int j = j0 + col;
        const float* gib = gi + (size_t)bidx * (3 * HID);
        const float r = sigmoidf_(gib[j] + aR[i] + bhh[j]);
        const float z = sigmoidf_(gib[HID + j] + aZ[i] + bhh[HID + j]);
        const float n = tanhf(gib[2 * HID + j] + r * (aN[i] + bhh[2 * HID + j]));
        const float hp = hprev[(size_t)bidx * HID + j];
        const float hn = (1.0f - z) * n + z * hp;
        hnext[(size_t)bidx * HID + j] = hn;
        dout[((size_t)bidx * LD + t) * HID + j] = hn;
    }
}

// ============================================================================
// K3: Y = relu(X @ W^T + bias), X:(M,512) row-major, W:(256,512), Y:(M,256)
//     2x2 register-blocked: one wave -> 32x32 output block.
// ============================================================================
__global__ __launch_bounds__(32) void mlp_kernel(
    const float* __restrict__ X,
    const float* __restrict__ W,
    const float* __restrict__ bias,
    float* __restrict__ Y)
{
    const int lane = threadIdx.x & 31;
    const int n0 = blockIdx.x * 32;
    const int m0 = blockIdx.y * 32;
    const int mr = lane & 15;
    const int k0 = (lane >> 4) << 1;

    const float* arow0 = X + (size_t)(m0 + mr) * 512;
    const float* arow1 = X + (size_t)(m0 + 16 + mr) * 512;
    const float* brow0 = W + (size_t)(n0 + mr) * 512;
    const float* brow1 = W + (size_t)(n0 + 16 + mr) * 512;

    v8f c00 = {}, c01 = {}, c10 = {}, c11 = {};
    for (int k = 0; k < 512; k += 4) {
        const v2f a0 = ld2(arow0 + k + k0);
        const v2f a1 = ld2(arow1 + k + k0);
        const v2f b0 = ld2(brow0 + k + k0);
        const v2f b1 = ld2(brow1 + k + k0);
        c00 = wmma_f32(a0, b0, c00);
        c01 = wmma_f32(a0, b1, c01);
        c10 = wmma_f32(a1, b0, c10);
        c11 = wmma_f32(a1, b1, c11);
    }
#pragma unroll
    for (int i = 0; i < 8; ++i) {
        const int row = i + ((lane >> 4) << 3);
        const int col = lane & 15;
        const int n_lo = n0 + col, n_hi = n0 + 16 + col;
        float* y0 = Y + (size_t)(m0 + row) * MLP;
        float* y1 = Y + (size_t)(m0 + 16 + row) * MLP;
        float v;
        v = c00[i] + bias[n_lo]; y0[n_lo] = v > 0.0f ? v : 0.0f;
        v = c01[i] + bias[n_hi]; y0[n_hi] = v > 0.0f ? v : 0.0f;
        v = c10[i] + bias[n_lo]; y1[n_lo] = v > 0.0f ? v : 0.0f;
        v = c11[i] + bias[n_hi]; y1[n_hi] = v > 0.0f ? v : 0.0f;
    }
}

// ============================================================================
// K4: outv[b][l][e] = dot(W[l], H[b][e][:])   (for out_e and out_d terms)
// ============================================================================
__global__ void bilinear_kernel(
    const float* __restrict__ H,   // (B, L, MLP)
    const float* __restrict__ W,   // (NL, MLP)
    float* __restrict__ outv,      // (B, NL, L)
    int L)
{
    const int idx = blockIdx.x * blockDim.x + threadIdx.x;
    const int total = B * NL * L;
    if (idx >= total) return;
    const int e = idx % L;
    const int l = (idx / L) % NL;
    const int b = idx / (L * NL);
    const float* h = H + ((size_t)b * L + e) * MLP;
    const float* w = W + (size_t)l * MLP;
    float s = 0.0f;
    for (int k = 0; k < MLP; ++k) s += w[k] * h[k];
    outv[idx] = s;
}

// ============================================================================
// K5: biaffine core. One workgroup (32 waves) per (b,l) pair.
//     T = d_h[b] @ U[l]  computed 256x32 k-chunk at a time into LDS,
//     S = T @ e_h[b]^T accumulated in registers across chunks.
//     Phase 2 blocks two d-tiles per e-tile so each (uncoalesced) e_h
//     fragment load feeds two WMMAs.
// ============================================================================
constexpr int KC = 32;       // k-chunk width
constexpr int TPAD = 258;    // LDS d-stride (256 + 2 pad: spreads banks)

__global__ __launch_bounds__(1024) void biaffine_kernel(
    const float* __restrict__ dh,    // (B, LD, MLP)
    const float* __restrict__ eh,    // (B, LE, MLP)
    const float* __restrict__ U,     // (NL, MLP, MLP)
    const float* __restrict__ oute,  // (B, NL, LE)
    const float* __restrict__ outd,  // (B, NL, LD)
    const float* __restrict__ baff,  // (NL)
    float* __restrict__ out)         // (B, LD, LE, NL)
{
    __shared__ float Tc[KC * TPAD];  // k-major: Tc[k*TPAD + d], 33 KB

    const int bidx = blockIdx.x >> 2;
    const int l    = blockIdx.x & 3;
    const int w    = threadIdx.x >> 5;
    const int lane = threadIdx.x & 31;
    const int mr = lane & 15;
    const int k0 = (lane >> 4) << 1;

    const float* dhb = dh + (size_t)bidx * LD * MLP;
    const float* ehb = eh + (size_t)bidx * LE * MLP;
    const float* Ul  = U + (size_t)l * MLP * MLP;

    // phase-1 tile per wave: 16 d-tiles x 2 k-tiles
    const int p1_d0  = (w >> 1) * 16;
    const int p1_kk0 = (w & 1) * 16;

    // phase-2 groups: 8 (32-wide d-groups) x 17 (e-tiles) = 136 groups,
    // group g = w + 32*j ; each group = two d-tiles sharing one B fragment.
    v8f acc[5][2];
#pragma unroll
    for (int j = 0; j < 5; ++j) { acc[j][0] = (v8f){}; acc[j][1] = (v8f){}; }

    for (int kc = 0; kc < MLP; kc += KC) {
        if (kc) __syncthreads();   // previous chunk fully consumed

        // ---- phase 1: T_chunk[d][kk] = sum_h d_h[b][d][h] * U[l][h][kc+kk]
        {
            const float* arow = dhb + (size_t)(p1_d0 + mr) * MLP;
            const int ncol = kc + p1_kk0 + mr;     // lane's U column
            v8f t = {};
            for (int h = 0; h < MLP; h += 4) {
                v2f a = ld2(arow + h + k0);
                v2f b;
                b.x = Ul[(size_t)(h + k0) * MLP + ncol];
                b.y = Ul[(size_t)(h + k0 + 1) * MLP + ncol];
                t = wmma_f32(a, b, t);
            }
#pragma unroll
            for (int i = 0; i < 8; ++i) {
                const int row = i + ((lane >> 4) << 3);   // d within tile
                const int col = lane & 15;                // kk within tile
                Tc[(p1_kk0 + col) * TPAD + (p1_d0 + row)] = t[i];
            }
        }
        __syncthreads();

        // ---- phase 2: S[d][e] += T_chunk[d][:] . e_h[b][e][kc:kc+KC]
#pragma unroll
        for (int j = 0; j < 5; ++j) {
            const int g = w + 32 * j;
            if (g >= 8 * 17) break;
            const int d0 = (g & 7) * 32;        // two d-tiles: d0, d0+16
            const int e0 = (g >> 3) * 16;
            int ecol = e0 + mr; if (ecol > LE - 1) ecol = LE - 1;  // clamp OOB
            const float* brow = ehb + (size_t)ecol * MLP + kc;
            v8f s0 = acc[j][0], s1 = acc[j][1];
            for (int kk = 0; kk < KC; kk += 4) {
                const v2f b = ld2(brow + kk + k0);
                v2f a0, a1;
                a0.x = Tc[(kk + k0) * TPAD + d0 + mr];
                a0.y = Tc[(kk + k0 + 1) * TPAD + d0 + mr];
                a1.x = Tc[(kk + k0) * TPAD + d0 + 16 + mr];
                a1.y = Tc[(kk + k0 + 1) * TPAD + d0 + 16 + mr];
                s0 = wmma_f32(a0, b, s0);
                s1 = wmma_f32(a1, b, s1);
            }
            acc[j][0] = s0; acc[j][1] = s1;
        }
    }

    // ---- epilogue: + out_d[b,l,d] + out_e[b,l,e] + b_aff[l], strided store
    const float bias_l = baff[l];
    const float* oute_bl = oute + ((size_t)bidx * NL + l) * LE;
    const float* outd_bl = outd + ((size_t)bidx * NL + l) * LD;
#pragma unroll
    for (int j = 0; j < 5; ++j) {
        const int g = w + 32 * j;
        if (g >= 8 * 17) break;
        const int d0g = (g & 7) * 32;
        const int e0  = (g >> 3) * 16;
#pragma unroll
        for (int half = 0; half < 2; ++half) {
            const int d0 = d0g + 16 * half;
#pragma unroll
            for (int i = 0; i < 8; ++i) {
                const int row = d0 + i + ((lane >> 4) << 3);
                const int col = e0 + (lane & 15);
                if (col < LE) {
                    const float v = acc[j][half][i] + outd_bl[row] + oute_bl[col] + bias_l;
                    out[(((size_t)bidx * LD + row) * LE + col) * NL + l] = v;
                }
            }
        }
    }
}

// ============================================================================
extern "C" void kernel_launch(void* const* d_in, const int* in_sizes, int n_in,
                              void* d_out, int out_size, void* d_ws, size_t ws_size,
                              hipStream_t stream) {
    const float* e_outputs = (const float*)d_in[0];
    const float* d_inputs  = (const float*)d_in[1];
    const float* h0        = (const float*)d_in[2];
    const float* W_ih      = (const float*)d_in[3];
    const float* W_hh      = (const float*)d_in[4];
    const float* b_ih      = (const float*)d_in[5];
    const float* b_hh      = (const float*)d_in[6];
    const float* e_mlp_w   = (const float*)d_in[7];
    const float* e_mlp_b   = (const float*)d_in[8];
    const float* d_mlp_w   = (const float*)d_in[9];
    const float* d_mlp_b   = (const float*)d_in[10];
    const float* W_e       = (const float*)d_in[11];
    const float* W_d       = (const float*)d_in[12];
    const float* U         = (const float*)d_in[13];
    const float* b_aff     = (const float*)d_in[14];
    float* out = (float*)d_out;

    // workspace layout (floats)
    float* ws = (float*)d_ws;
    size_t off = 0;
    float* GI    = ws + off; off += (size_t)LD * B * 3 * HID;  // 50.3 MB
    float* hbuf0 = ws + off; off += (size_t)B * HID;
    float* hbuf1 = ws + off; off += (size_t)B * HID;
    float* DOUT  = ws + off; off += (size_t)B * LD * HID;      // 16.8 MB
    float* EH    = ws + off; off += (size_t)B * LE * MLP;      // 8.5 MB
    float* DH    = ws + off; off += (size_t)B * LD * MLP;      // 8.4 MB
    float* OUTE  = ws + off; off += (size_t)B * NL * LE;
    float* OUTD  = ws + off; off += (size_t)B * NL * LD;
    (void)ws_size; (void)in_sizes; (void)n_in; (void)out_size;

    // 1) precompute input gates for all steps (parallel WMMA GEMM)
    gi_kernel<<<dim3((3 * HID) / 32, LD), 32, 0, stream>>>(
        d_inputs, W_ih, b_ih, GI);

    // 2) sequential GRU: h ping-pong through workspace
    hipMemcpyAsync(hbuf0, h0, (size_t)B * HID * sizeof(float),
                   hipMemcpyDeviceToDevice, stream);
    float* bufs[2] = {hbuf0, hbuf1};
    for (int t = 0; t < LD; ++t) {
        gru_step_kernel<<<dim3(HID / 16, B / 16), 32, 0, stream>>>(
            bufs[t & 1], W_hh, b_hh, GI, bufs[(t + 1) & 1], DOUT, t);
    }

    // 3) MLP projections (WMMA GEMM + bias + relu)
    mlp_kernel<<<dim3(MLP / 32, (B * LE) / 32), 32, 0, stream>>>(
        e_outputs, e_mlp_w, e_mlp_b, EH);
    mlp_kernel<<<dim3(MLP / 32, (B * LD) / 32), 32, 0, stream>>>(
        DOUT, d_mlp_w, d_mlp_b, DH);

    // 4) linear bias terms out_e / out_d
    {
        const int totE = B * NL * LE;
        bilinear_kernel<<<(totE + 255) / 256, 256, 0, stream>>>(EH, W_e, OUTE, LE);
        const int totD = B * NL * LD;
        bilinear_kernel<<<(totD + 255) / 256, 256, 0, stream>>>(DH, W_d, OUTD, LD);
    }

    // 5) biaffine core: one workgroup per (b,l)
    biaffine_kernel<<<B * NL, 1024, 0, stream>>>(
        DH, EH, U, OUTE, OUTD, b_aff, out);
}